// TransformerBlockQuantum_65481071402759
// MI455X (gfx1250) — compile-verified
//
#include <hip/hip_runtime.h>

#define EPS 1e-5f
#define ROWS_PER_BLOCK 256   // 8 waves x 32 rows, one row per thread

typedef __attribute__((ext_vector_type(2))) float v2f;
typedef __attribute__((ext_vector_type(8))) float v8f;

// Fused: LN1 -> analytic quantum layer -> (z @ Wc^T + bc) via WMMA -> LN2
// Quantum layer closed form:
//   c_w = cos(h_w + theta_w)
//   z_j = prod_{w=0..j} c_w   (j >= 1)
//   z_0 = prod_{w=1..7} c_w
__global__ __launch_bounds__(ROWS_PER_BLOCK)
void tbq_fused_kernel(const float* __restrict__ x,
                      const float* __restrict__ gamma1, const float* __restrict__ beta1,
                      const float* __restrict__ theta,
                      const float* __restrict__ Wc,     const float* __restrict__ bc,
                      const float* __restrict__ gamma2, const float* __restrict__ beta2,
                      float* __restrict__ out, int nrows)
{
    // pad row stride to 9 words -> conflict-free LDS column access (gcd(9,64)=1)
    __shared__ float zbuf[ROWS_PER_BLOCK][9];

    const int tid      = threadIdx.x;
    const int lane     = tid & 31;
    const int half     = lane >> 4;        // which 16-lane half of the wave
    const int l16      = lane & 15;
    const int waveBase = tid & ~31;        // block-local first row of this wave

    size_t row = (size_t)blockIdx.x * ROWS_PER_BLOCK + tid;
    const size_t rowc = (row < (size_t)nrows) ? row : (size_t)(nrows - 1); // clamp loads

    // ---------------- LN1 + quantum layer (per-thread, E=8) ----------------
    const float4 lo = *(const float4*)(x + rowc * 8);
    const float4 hi = *(const float4*)(x + rowc * 8 + 4);
    float v[8] = {lo.x, lo.y, lo.z, lo.w, hi.x, hi.y, hi.z, hi.w};

    float m = 0.f;
    #pragma unroll
    for (int i = 0; i < 8; i++) m += v[i];
    m *= 0.125f;
    float var = 0.f;
    #pragma unroll
    for (int i = 0; i < 8; i++) { float d = v[i] - m; var += d * d; }
    var *= 0.125f;
    const float rs = rsqrtf(var + EPS);

    float c[8];
    #pragma unroll
    for (int i = 0; i < 8; i++) {
        float h = (v[i] - m) * rs * gamma1[i] + beta1[i];
        c[i] = __cosf(h + theta[i]);            // v_cos_f32; |arg| is small here
    }
    float z[8];
    float p = c[0];
    #pragma unroll
    for (int j = 1; j < 8; j++) { p *= c[j]; z[j] = p; }   // prefix products
    float q = c[1];
    #pragma unroll
    for (int j = 2; j < 8; j++) q *= c[j];
    z[0] = q;

    #pragma unroll
    for (int i = 0; i < 8; i++) zbuf[tid][i] = z[i];
    __syncthreads();

    // ------------- projection via V_WMMA_F32_16X16X4_F32, 2 tiles/wave -------------
    // B (4x16, K x N), N>=8 zero-padded: v0 = {K=kb | K=kb} over half-waves, kb = 2*half
    const int n  = l16;          // output column this lane owns in D
    const int kb = half * 2;     // K pair handled by this half-wave (0,1) or (2,3)
    v2f B0; B0[0] = 0.f; B0[1] = 0.f;   // K-block 0..3
    v2f B1; B1[0] = 0.f; B1[1] = 0.f;   // K-block 4..7
    float bcn = 0.f;
    if (n < 8) {
        B0[0] = Wc[n * 8 + kb];       // B[k][n] = Wc[n][k]
        B0[1] = Wc[n * 8 + kb + 1];
        B1[0] = Wc[n * 8 + 4 + kb];
        B1[1] = Wc[n * 8 + 4 + kb + 1];
        bcn   = bc[n];
    }

    #pragma unroll
    for (int t = 0; t < 2; t++) {
        // A (16x4): lane l16 holds row (tileBase + l16), half-wave picks K pair kb
        const int arow = waveBase + t * 16 + l16;
        v2f A0, A1;
        A0[0] = zbuf[arow][kb];       A0[1] = zbuf[arow][kb + 1];
        A1[0] = zbuf[arow][4 + kb];   A1[1] = zbuf[arow][4 + kb + 1];

        v8f acc = {};
        acc = __builtin_amdgcn_wmma_f32_16x16x4_f32(false, A0, false, B0,
                                                    (short)0, acc, false, false);
        acc = __builtin_amdgcn_wmma_f32_16x16x4_f32(false, A1, false, B1,
                                                    (short)0, acc, false, false);

        // D: vgpr r -> (M = r + 8*half, N = l16). Fold bc into write-back.
        // Tile0 overwrites rows 0..15 only; tile1's A-reads touch rows 16..31,
        // and same-wave DS ops are in-order, so reuse of zbuf is safe.
        if (n < 8) {
            const int mbase = waveBase + t * 16 + half * 8;
            #pragma unroll
            for (int r = 0; r < 8; r++)
                zbuf[mbase + r][n] = acc[r] + bcn;
        }
    }
    __syncthreads();

    // ---------------------------- LN2 + store ----------------------------
    float o[8];
    #pragma unroll
    for (int i = 0; i < 8; i++) o[i] = zbuf[tid][i];
    float m2 = 0.f;
    #pragma unroll
    for (int i = 0; i < 8; i++) m2 += o[i];
    m2 *= 0.125f;
    float var2 = 0.f;
    #pragma unroll
    for (int i = 0; i < 8; i++) { float d = o[i] - m2; var2 += d * d; }
    var2 *= 0.125f;
    const float rs2 = rsqrtf(var2 + EPS);

    if (row < (size_t)nrows) {
        float4 r0, r1;
        r0.x = (o[0]-m2)*rs2*gamma2[0] + beta2[0];
        r0.y = (o[1]-m2)*rs2*gamma2[1] + beta2[1];
        r0.z = (o[2]-m2)*rs2*gamma2[2] + beta2[2];
        r0.w = (o[3]-m2)*rs2*gamma2[3] + beta2[3];
        r1.x = (o[4]-m2)*rs2*gamma2[4] + beta2[4];
        r1.y = (o[5]-m2)*rs2*gamma2[5] + beta2[5];
        r1.z = (o[6]-m2)*rs2*gamma2[6] + beta2[6];
        r1.w = (o[7]-m2)*rs2*gamma2[7] + beta2[7];
        *(float4*)(out + row * 8)     = r0;
        *(float4*)(out + row * 8 + 4) = r1;
    }
}

extern "C" void kernel_launch(void* const* d_in, const int* in_sizes, int n_in,
                              void* d_out, int out_size, void* d_ws, size_t ws_size,
                              hipStream_t stream) {
    (void)n_in; (void)d_ws; (void)ws_size; (void)out_size;
    const float* x      = (const float*)d_in[0];
    const float* gamma1 = (const float*)d_in[1];
    const float* beta1  = (const float*)d_in[2];
    const float* theta  = (const float*)d_in[3];
    const float* Wc     = (const float*)d_in[4];
    const float* bc     = (const float*)d_in[5];
    const float* gamma2 = (const float*)d_in[6];
    const float* beta2  = (const float*)d_in[7];
    float* out = (float*)d_out;

    const int nrows  = in_sizes[0] / 8;                       // B*S = 131072
    const int blocks = (nrows + ROWS_PER_BLOCK - 1) / ROWS_PER_BLOCK;
    tbq_fused_kernel<<<blocks, ROWS_PER_BLOCK, 0, stream>>>(
        x, gamma1, beta1, theta, Wc, bc, gamma2, beta2, out, nrows);
}